// PDETestError_42030549958923
// MI455X (gfx1250) — compile-verified
//
#include <hip/hip_runtime.h>

typedef __attribute__((ext_vector_type(2))) float v2f;
typedef __attribute__((ext_vector_type(8))) float v8f;

#define TILE 256       // M*N = 16*16
#define NDER 5
#define NB   16        // batch
#define NS   1024      // inner patches

// workspace layout in floats
#define WS_X 0
#define WS_Y (NB*4096)
#define WS_Q (2*NB*4096)
#define WS_PART (3*NB*4096)

// ---------------------------------------------------------------------------
// Kernel 1: gather control points + masked coefficients
// ---------------------------------------------------------------------------
__global__ void pde_gather_kernel(const float* __restrict__ inp,
                                  const float* __restrict__ outp,
                                  const int* __restrict__ mapRecord,
                                  float* __restrict__ ws) {
    int i = blockIdx.x * blockDim.x + threadIdx.x;   // 0..65535
    int b   = i >> 12;                               // / 4096
    int r = mapRecord[2 * i + 0];
    int c = mapRecord[2 * i + 1];
    int rc = r * 64 + c;
    float w  = inp[(b * 3 + 0) * 4096 + rc];
    float cx = inp[(b * 3 + 1) * 4096 + rc];
    float cy = inp[(b * 3 + 2) * 4096 + rc];
    float o  = outp[b * 4096 + rc];
    float cf = (w > 0.0f) ? o : 0.0f;
    ws[WS_X + i] = cx;
    ws[WS_Y + i] = cy;
    ws[WS_Q + i] = cf;
}

// ---------------------------------------------------------------------------
// Kernel 2: per-s WMMA GEMMs (U = X@D^T, V = Y@D^T, T = Q@D^T) + epilogue
// one wave32 per s
// ---------------------------------------------------------------------------
__global__ __launch_bounds__(32) void pde_main_kernel(
        const float* __restrict__ ders,        // (1024,5,16,16)
        const int* __restrict__ inner_patch,   // (1024,)
        const float* __restrict__ ws_in,
        float* __restrict__ partial) {
    __shared__ float sD[NDER * TILE];
    __shared__ float sX[NB * TILE];
    __shared__ float sY[NB * TILE];
    __shared__ float sQ[NB * TILE];
    __shared__ float sU[256];
    __shared__ float sV[256];
    __shared__ float sT[256];
    __shared__ float sRed[16];

    const int s   = blockIdx.x;
    const int tid = threadIdx.x;
    const int p   = inner_patch[s];

    // ---- stage D (5x256) ----
    {
        const float4* g = (const float4*)(ders + (size_t)s * NDER * TILE);
        float4* d = (float4*)sD;
        for (int i = tid; i < NDER * TILE / 4; i += 32) d[i] = g[i];
    }
    // ---- stage selected X/Y/Q rows (16 x 256 each) ----
    {
        const float* Xw = ws_in + WS_X;
        const float* Yw = ws_in + WS_Y;
        const float* Qw = ws_in + WS_Q;
        for (int b = 0; b < NB; ++b) {
            int off = b * 4096 + p * TILE;
            const float4* sx = (const float4*)(Xw + off);
            const float4* sy = (const float4*)(Yw + off);
            const float4* sq = (const float4*)(Qw + off);
            float4* dx = (float4*)(sX + b * TILE);
            float4* dy = (float4*)(sY + b * TILE);
            float4* dq = (float4*)(sQ + b * TILE);
            for (int i = tid; i < TILE / 4; i += 32) {
                dx[i] = sx[i]; dy[i] = sy[i]; dq[i] = sq[i];
            }
        }
    }
    __syncthreads();

    // ---- WMMA f32 16x16x4 fragment addressing ----
    // A (16x4): lane m = tid&15 holds K = 4*kk + koff + {0,1}, koff = 2*(tid>=16)
    // B (4x16): lane n = tid&15 holds same K pair from row n of D (n>=5 -> 0)
    const int m    = tid & 15;
    const int koff = (tid >> 4) << 1;
    const int mrow = (m < NDER) ? m : 0;
    const float msk = (m < NDER) ? 1.0f : 0.0f;
    const float* aXp = sX + m * TILE + koff;
    const float* aYp = sY + m * TILE + koff;
    const float* aQp = sQ + m * TILE + koff;
    const float* bDp = sD + mrow * TILE + koff;

    v8f accU = {}; v8f accV = {}; v8f accT = {};
    #pragma unroll 8
    for (int kk = 0; kk < 64; ++kk) {
        const int kb = kk * 4;
        v2f bD = *(const v2f*)(bDp + kb);
        bD *= msk;                       // zero-pad der rows 5..15, no divergence
        v2f aX = *(const v2f*)(aXp + kb);
        v2f aY = *(const v2f*)(aYp + kb);
        v2f aQ = *(const v2f*)(aQp + kb);
        accU = __builtin_amdgcn_wmma_f32_16x16x4_f32(false, aX, false, bD, (short)0, accU, false, false);
        accV = __builtin_amdgcn_wmma_f32_16x16x4_f32(false, aY, false, bD, (short)0, accV, false, false);
        accT = __builtin_amdgcn_wmma_f32_16x16x4_f32(false, aQ, false, bD, (short)0, accT, false, false);
    }

    // ---- spill C matrices to LDS (C layout: vgpr v -> row v + 8*(lane>=16), col = lane&15) ----
    {
        const int col = tid & 15;
        const int rb  = (tid >> 4) * 8;
        #pragma unroll
        for (int v = 0; v < 8; ++v) {
            sU[(rb + v) * 16 + col] = accU[v];
            sV[(rb + v) * 16 + col] = accV[v];
            sT[(rb + v) * 16 + col] = accT[v];
        }
    }
    __syncthreads();

    // ---- scalar epilogue: lanes 0..15 handle b = lane ----
    if (tid < 16) {
        const float* Ur = sU + tid * 16;
        const float* Vr = sV + tid * 16;
        const float* Tr = sT + tid * 16;
        float s0 = Ur[0], s2 = Ur[1], s4 = Ur[2], s6 = Ur[3], s8 = Ur[4];
        float s1 = Vr[0], s3 = Vr[1], s5 = Vr[2], s7 = Vr[3], s9 = Vr[4];
        float t0 = Tr[0], t1 = Tr[1], t2 = Tr[2], t3 = Tr[3], t4 = Tr[4];

        float J2  = s0 * s3 - s1 * s2;
        float a00 = s0 * s0, a01 = 2.0f * s0 * s1, a02 = s1 * s1;
        float a10 = s0 * s2, a11 = s0 * s3 + s2 * s1, a12 = s1 * s3;
        float a20 = s2 * s2, a21 = 2.0f * s2 * s3, a22 = s3 * s3;
        float J3  = a00 * a11 * a22 + a01 * a12 * a20 + a02 * a10 * a21
                  - a02 * a11 * a20 - a00 * a12 * a21 - a01 * a10 * a22;
        float A_ = a11 * a22 - a21 * a12, D_ = a10 * a21 - a20 * a11;
        float B_ = a21 * a02 - a01 * a22, E_ = a20 * a01 - a00 * a21;
        float C_ = a01 * a12 - a02 * a11, F_ = a00 * a11 - a01 * a10;
        float inv3 = 1.0f / J3;
        float G1 = (A_ + D_) * inv3, G2 = (B_ + E_) * inv3, G3 = (C_ + F_) * inv3;
        float P  = G1 * s4 + G2 * s8 + G3 * s6;
        float Qc = G1 * s5 + G2 * s9 + G3 * s7;
        float invJ2 = 1.0f / J2;
        float al0 = (Qc * s2 - P * s3) * invJ2;
        float al1 = (P * s1 - Qc * s0) * invJ2;
        float resid = al0 * t0 + al1 * t1 + G1 * t2 + G3 * t3 + G2 * t4 + 100.0f;
        sRed[tid] = fabsf(resid);
    }
    __syncthreads();
    if (tid == 0) {
        float acc = 0.0f;
        #pragma unroll
        for (int i = 0; i < 16; ++i) acc += sRed[i];   // fixed order -> deterministic
        partial[s] = acc;
    }
}

// ---------------------------------------------------------------------------
// Kernel 3: deterministic final reduction of 1024 partials -> mean
// ---------------------------------------------------------------------------
__global__ __launch_bounds__(32) void pde_reduce_kernel(const float* __restrict__ partial,
                                                        float* __restrict__ out) {
    __shared__ float sh[32];
    int t = threadIdx.x;
    float a = 0.0f;
    for (int j = 0; j < 32; ++j) a += partial[t + 32 * j];  // fixed order
    sh[t] = a;
    __syncthreads();
    if (t == 0) {
        float tot = 0.0f;
        for (int i = 0; i < 32; ++i) tot += sh[i];
        out[0] = tot * (1.0f / 16384.0f);                   // mean over B*S
    }
}

extern "C" void kernel_launch(void* const* d_in, const int* in_sizes, int n_in,
                              void* d_out, int out_size, void* d_ws, size_t ws_size,
                              hipStream_t stream) {
    const float* inp         = (const float*)d_in[0];  // (16,3,64,64)
    const float* outp        = (const float*)d_in[1];  // (16,1,64,64)
    const float* ders        = (const float*)d_in[2];  // (1024,5,16,16)
    const int*   mapRecord   = (const int*)d_in[3];    // (16,4096,2)
    const int*   inner_patch = (const int*)d_in[4];    // (1024,)
    float* ws  = (float*)d_ws;
    float* out = (float*)d_out;

    pde_gather_kernel<<<256, 256, 0, stream>>>(inp, outp, mapRecord, ws);
    pde_main_kernel<<<NS, 32, 0, stream>>>(ders, inner_patch, ws, ws + WS_PART);
    pde_reduce_kernel<<<1, 32, 0, stream>>>(ws + WS_PART, out);
}